// CRF_12171937317521
// MI455X (gfx1250) — compile-verified
//
#include <hip/hip_runtime.h>
#include <stdint.h>

#define B_  256
#define T_  1024
#define C_  128
#define CH_ 64   // backward chunk: rows of backpointers staged in LDS

typedef unsigned int u32x4 __attribute__((ext_vector_type(4)));
typedef int          i32x8 __attribute__((ext_vector_type(8)));
typedef int          i32x4 __attribute__((ext_vector_type(4)));

// One block per batch element. blockDim.x == C_ (lane == tag c), 4 wave32s.
__global__ __launch_bounds__(C_) void viterbi_crf_kernel(
    const float* __restrict__ pot,     // [B,T,C]
    const float* __restrict__ trans,   // [C,C]  (prev p -> cur c)
    float* __restrict__ out)           // [B,T,C] one-hot
{
  __shared__ float s_trans[C_ * C_];              // 64 KB, layout [p][c]
  __shared__ float s_alphaA[C_];
  __shared__ float s_alphaB[C_];
  __shared__ int   s_tags[CH_];
  __shared__ int   s_last;
  __shared__ __align__(16) unsigned char s_bp[CH_ * C_];   // 8 KB chunk of backpointers

  const int c = threadIdx.x;
  const int b = blockIdx.x;

  // ---------- Stage transitions [C,C] f32 (64 KB) into LDS ----------
#if __has_builtin(__builtin_amdgcn_tensor_load_to_lds) && __has_builtin(__builtin_amdgcn_s_wait_tensorcnt)
  if (threadIdx.x < 32) {   // wave 0 issues one TDM 2-D tile load (EXEC ignored by TDM)
    uint64_t ga = (uint64_t)(uintptr_t)trans;
    uint32_t la = (uint32_t)(uintptr_t)s_trans;   // generic LDS ptr low 32 bits = LDS byte offset
    u32x4 g0;
    g0[0] = 1u;                                     // count=1 (valid descriptor), user mode
    g0[1] = la;                                     // lds_addr
    g0[2] = (uint32_t)ga;                           // global_addr[31:0]
    g0[3] = (uint32_t)((ga >> 32) & 0x01FFFFFFu)    // global_addr[56:32]
            | 0x80000000u;                          // type = 2 ("image")
    i32x8 g1;
    g1[0] = 0x00020000;            // workgroup_mask=0, data_size=2 (4 bytes)
    g1[1] = (int)(128u << 16);     // tensor_dim0 = 128 (bits 79:48)
    g1[2] = (int)(128u << 16);     // tensor_dim1 = 128 (bits 111:80)
    g1[3] = (int)(128u << 16);     // tile_dim0   = 128 (bits 127:112)
    g1[4] = 128;                   // tile_dim1 = 128, tile_dim2 = 0
    g1[5] = 128;                   // tensor_dim0_stride = 128
    g1[6] = 0;                     // stride0 hi / stride1 lo
    g1[7] = 0;                     // stride1 hi
    i32x4 gz4 = {0, 0, 0, 0};              // 2-D tensor: groups 2/3 unused
    i32x8 gz8 = {0, 0, 0, 0, 0, 0, 0, 0};  // extra group (clang-23 6-arg form)
    __builtin_amdgcn_tensor_load_to_lds(g0, g1, gz4, gz4, gz8, 0);
    __builtin_amdgcn_s_wait_tensorcnt(0);
  }
#else
  for (int i = threadIdx.x; i < C_ * C_ / 4; i += C_)
    ((float4*)s_trans)[i] = ((const float4*)trans)[i];
#endif

  // ---------- Forward max-plus DP ----------
  const float* potb = pot + (size_t)b * T_ * C_ + c;   // this lane's column
  unsigned char* bpBase = (unsigned char*)out;         // bp row t lives in first 128B of out row (b,t)

  s_alphaA[c] = potb[0];      // alpha0 = pots[b,0,:]
  __syncthreads();            // covers both the trans staging and alpha0

  float* aCur = s_alphaA;
  float* aNxt = s_alphaB;

  for (int t = 1; t < T_; ++t) {
    float pv = potb[(size_t)t * C_];                 // unary potential for (b,t,c)
    if (t + 16 < T_)
      __builtin_prefetch(potb + (size_t)(t + 16) * C_, 0, 0);  // global_prefetch_b8

    float best = -3.402823466e38f;
    int   arg  = 0;
#pragma unroll 8
    for (int p = 0; p < C_; ++p) {
      // aCur[p]: uniform LDS broadcast; s_trans[p*C_+c]: stride-1 across lanes (no bank conflicts)
      float s = aCur[p] + s_trans[p * C_ + c];
      if (s > best) { best = s; arg = p; }           // strict '>' => first max index (jnp.argmax)
    }
    aNxt[c] = best + pv;
    bpBase[((size_t)b * T_ + t) * (C_ * 4) + c] = (unsigned char)arg;   // coalesced 128B row
    __syncthreads();
    float* tmp = aCur; aCur = aNxt; aNxt = tmp;
  }

  // last tag = argmax_c alpha_T (first max index)
  if (c == 0) {
    float bv = aCur[0]; int bi = 0;
    for (int i = 1; i < C_; ++i) { float v = aCur[i]; if (v > bv) { bv = v; bi = i; } }
    s_last = bi;
  }
  __threadfence_block();   // make forward bp stores visible before async re-reads
  __syncthreads();

  // ---------- Backward trace + one-hot emit, chunked through LDS ----------
  int cur = s_last;        // only thread 0's copy is authoritative across chunks

  for (int tHi = T_ - 1; tHi >= 0; tHi -= CH_) {
    int tLo    = tHi - CH_ + 1; if (tLo < 0) tLo = 0;
    int tFirst = (tLo > 0) ? tLo : 1;                // bp rows exist for t >= 1
    int nrows  = tHi - tFirst + 1;

    // Async-copy bp rows tFirst..tHi (128B each) into LDS slots (t - tLo).
    const uint32_t lbase = (uint32_t)(uintptr_t)s_bp;
    for (int i = threadIdx.x; i < nrows * (C_ / 16); i += C_) {
      int r = i >> 3, w = i & 7;                     // row, 16B word within row
      int t = tFirst + r;
      uint64_t ga = (uint64_t)(uintptr_t)(bpBase + ((size_t)b * T_ + t) * (C_ * 4) + w * 16);
      uint32_t la = lbase + (uint32_t)((t - tLo) * C_ + w * 16);
      asm volatile("global_load_async_to_lds_b128 %0, %1, off"
                   :: "v"(la), "v"(ga) : "memory");
    }
    asm volatile("s_wait_asynccnt 0" ::: "memory");
    __syncthreads();

    // Thread 0 walks the chain at LDS latency.
    if (c == 0) {
      int cc = cur;
      s_tags[tHi - tLo] = cc;
      for (int t = tHi; t > tLo; --t) {
        cc = s_bp[(t - tLo) * C_ + cc];              // tag[t-1] = bp[t][tag[t]]
        s_tags[t - 1 - tLo] = cc;
      }
      if (tLo > 0) cc = s_bp[cc];                    // pre-compute tag[tLo-1] before rows get overwritten
      cur = cc;
    }
    __syncthreads();

    // All lanes emit one-hot rows (coalesced 512B stores), overwriting the bp bytes.
    for (int t = tLo; t <= tHi; ++t)
      out[((size_t)b * T_ + t) * C_ + c] = (c == s_tags[t - tLo]) ? 1.0f : 0.0f;
  }
}

extern "C" void kernel_launch(void* const* d_in, const int* in_sizes, int n_in,
                              void* d_out, int out_size, void* d_ws, size_t ws_size,
                              hipStream_t stream) {
  (void)in_sizes; (void)n_in; (void)out_size; (void)d_ws; (void)ws_size;
  const float* pot   = (const float*)d_in[0];   // [B,T,C] f32
  const float* trans = (const float*)d_in[1];   // [C,C]   f32
  float*       out   = (float*)d_out;           // [B,T,C] f32 one-hot
  viterbi_crf_kernel<<<dim3(B_), dim3(C_), 0, stream>>>(pot, trans, out);
}